// KNNInterpolate_42090679501099
// MI455X (gfx1250) — compile-verified
//
#include <hip/hip_runtime.h>
#include <hip/hip_bf16.h>

// KNN feature interpolation, K=3, C=128, f32.
// Per 16-query group: D(16x128) = A(16x48, banded weights) x B(48x128, gathered
// feature rows), evaluated as 12 chained V_WMMA_F32_16X16X4_F32 per 16-col tile,
// 8 tiles fully unrolled so B gathers use immediate offsets off 24 fixed addresses.

typedef __attribute__((ext_vector_type(2))) float v2f;
typedef __attribute__((ext_vector_type(8))) float v8f;

#define KNN 3
#define CF 128
#define QPG 16                      // queries per wave-group (WMMA M dim)
#define WAVES_PER_BLOCK 8
#define THREADS (WAVES_PER_BLOCK * 32)
#define NKCHUNK 12                  // 48 / 4 (K chunks of the 16x16x4 f32 WMMA)

__global__ __launch_bounds__(THREADS)
void KNNInterpolate_wmma_kernel(const float* __restrict__ s_feats,
                                const float* __restrict__ q_points,
                                const float* __restrict__ s_points,
                                const int*   __restrict__ nn_idx,
                                float*       __restrict__ out,
                                int M)
{
    __shared__ unsigned rowelem[WAVES_PER_BLOCK][QPG * KNN];  // element offsets into s_feats

    const int lane = threadIdx.x & 31;
    const int wave = threadIdx.x >> 5;
    const int h    = lane >> 4;        // half-wave select
    const int mrow = lane & 15;        // row (A/D) or column (B/D) index in tile
    const int base = (blockIdx.x * WAVES_PER_BLOCK + wave) * QPG;

    // ---- per-query inverse-distance weights (lane's own row; both half-waves
    //      redundantly compute identical values, so no cross-lane traffic) ----
    const int q = min(base + mrow, M - 1);
    float w[KNN];
    float wsum = 0.0f;
    const float qx = q_points[q * 3 + 0];
    const float qy = q_points[q * 3 + 1];
    const float qz = q_points[q * 3 + 2];
    #pragma unroll
    for (int k = 0; k < KNN; ++k) {
        const int idx = nn_idx[q * KNN + k];
        const float dx = qx - s_points[idx * 3 + 0];
        const float dy = qy - s_points[idx * 3 + 1];
        const float dz = qz - s_points[idx * 3 + 2];
        const float d2 = dx * dx + dy * dy + dz * dz;
        w[k] = 1.0f / (d2 + 1e-8f);
        wsum += w[k];
        if (h == 0) rowelem[wave][mrow * KNN + k] = (unsigned)idx * (unsigned)CF;
    }
    const float winv = 1.0f / wsum;
    w[0] *= winv; w[1] *= winv; w[2] *= winv;

    __syncthreads();   // publish the 48 row offsets within the block

    // ---- hoist A-matrix fragments and per-lane gather base addresses ----
    // A layout (16x4 f32): lane L holds A[L&15, 2h+v] in VGPR v; global K = 4*kk + 2h + v.
    // rowp[kk][v] points at this lane's column (tile 0) in gathered row j; across
    // tiles the address only shifts by nt*64 bytes -> instruction IOFFSET.
    float        afrag[NKCHUNK][2];
    const float* rowp [NKCHUNK][2];
    #pragma unroll
    for (int kk = 0; kk < NKCHUNK; ++kk) {
        #pragma unroll
        for (int v = 0; v < 2; ++v) {
            const int j  = 4 * kk + 2 * h + v;      // 0..47 == m'*3 + k
            const int jm = j / KNN;
            const int jk = j - jm * KNN;
            afrag[kk][v] = (jm == mrow) ? w[jk] : 0.0f;
            rowp [kk][v] = s_feats + rowelem[wave][j] + (unsigned)mrow;
        }
    }

    // Scalar (wave-uniform) bounds check: base is identical across the wave.
    const int  sbase = __builtin_amdgcn_readfirstlane(base);
    const bool full  = (sbase + QPG) <= M;

    if (full) {
        // ---- fast path: EXEC all-ones; nt fully unrolled so every B load is
        //      global_load_b32 with an immediate offset off a fixed address ----
        float* __restrict__ op = out + (size_t)(sbase + 8 * h) * CF + mrow;
        #pragma unroll
        for (int nt = 0; nt < CF / 16; ++nt) {
            v8f acc = {};
            #pragma unroll
            for (int kk = 0; kk < NKCHUNK; ++kk) {
                v2f av, bv;
                av.x = afrag[kk][0];
                av.y = afrag[kk][1];
                // B layout (4x16 f32): lane L holds B[2h+v, L&15] in VGPR v.
                bv.x = rowp[kk][0][nt * 16];
                bv.y = rowp[kk][1][nt * 16];
                acc = __builtin_amdgcn_wmma_f32_16x16x4_f32(
                    false, av, false, bv, (short)0, acc, false, false);
            }
            // D layout: lane L, VGPR r -> D[r + 8h, L&15]
            #pragma unroll
            for (int r = 0; r < 8; ++r) {
                op[r * CF + nt * 16] = acc[r];
            }
        }
    } else {
        // ---- remainder path (only if M % 16 != 0 for the last group) ----
        #pragma unroll 1
        for (int nt = 0; nt < CF / 16; ++nt) {
            v8f acc = {};
            #pragma unroll
            for (int kk = 0; kk < NKCHUNK; ++kk) {
                v2f av, bv;
                av.x = afrag[kk][0];
                av.y = afrag[kk][1];
                bv.x = rowp[kk][0][nt * 16];
                bv.y = rowp[kk][1][nt * 16];
                acc = __builtin_amdgcn_wmma_f32_16x16x4_f32(
                    false, av, false, bv, (short)0, acc, false, false);
            }
            #pragma unroll
            for (int r = 0; r < 8; ++r) {
                const int row = sbase + r + 8 * h;
                if (row < M) out[(size_t)row * CF + nt * 16 + mrow] = acc[r];
            }
        }
    }
}

extern "C" void kernel_launch(void* const* d_in, const int* in_sizes, int n_in,
                              void* d_out, int out_size, void* d_ws, size_t ws_size,
                              hipStream_t stream) {
    const float* s_feats  = (const float*)d_in[0];   // [N_SRC, 128] f32
    const float* q_points = (const float*)d_in[1];   // [M, 3] f32
    const float* s_points = (const float*)d_in[2];   // [N_SRC, 3] f32
    const int*   nn_idx   = (const int*)  d_in[3];   // [M, 3] int (harness int mapping)
    float*       out      = (float*)d_out;           // [M, 128] f32

    const int M = in_sizes[1] / 3;                   // 262144
    const int queries_per_block = WAVES_PER_BLOCK * QPG;  // 128
    const int grid = (M + queries_per_block - 1) / queries_per_block;

    KNNInterpolate_wmma_kernel<<<grid, THREADS, 0, stream>>>(
        s_feats, q_points, s_points, nn_idx, out, M);
}